// GCNGraphClassifier_75642964017720
// MI455X (gfx1250) — compile-verified
//
#include <hip/hip_runtime.h>
#include <hip/hip_bf16.h>

typedef __attribute__((ext_vector_type(2))) float v2f;
typedef __attribute__((ext_vector_type(8))) float v8f;

#define HID 32
#define EPSN 1e-5f

// ---------------------------------------------------------------------------
// Degree / dinv
// ---------------------------------------------------------------------------
__global__ void k_init_deg(float* __restrict__ deg, int N) {
    int i = blockIdx.x * blockDim.x + threadIdx.x;
    if (i < N) deg[i] = 1.0f;  // self-loop contributes 1 to every node
}

__global__ void k_deg_edges(const int* __restrict__ col, float* __restrict__ deg, int E) {
    int e = blockIdx.x * blockDim.x + threadIdx.x;
    if (e < E) atomicAdd(&deg[col[e]], 1.0f);
}

__global__ void k_dinv(float* __restrict__ deg, int N) {
    int i = blockIdx.x * blockDim.x + threadIdx.x;
    if (i < N) deg[i] = rsqrtf(deg[i]);  // deg >= 1 always (self-loops)
}

// ---------------------------------------------------------------------------
// Per-graph node ranges from the sorted batch vector
// ---------------------------------------------------------------------------
__global__ void k_graph_offsets(const int* __restrict__ batch, int* __restrict__ off,
                                int N, int G) {
    int i = blockIdx.x * blockDim.x + threadIdx.x;
    if (i >= N) return;
    if (i == 0) {
        for (int g = 0; g <= batch[0]; ++g) off[g] = 0;
    } else if (batch[i] != batch[i - 1]) {
        for (int g = batch[i - 1] + 1; g <= batch[i]; ++g) off[g] = i;
    }
    if (i == N - 1) {
        for (int g = batch[N - 1] + 1; g <= G; ++g) off[g] = N;
    }
}

// ---------------------------------------------------------------------------
// WMMA GEMM: H[N,32] = X[N,K] @ W[K,32]   (fp32 via V_WMMA_F32_16X16X4_F32)
// A layout (16x4 f32): lane = (k/2)*16 + m, vgpr = k%2
// B layout (4x16 f32): lane = (k/2)*16 + n, vgpr = k%2
// D layout (16x16 f32): vgpr r -> M = r + 8*(lane>=16), N = lane&15
// One wave computes a 16-row x 32-col tile (two 16x16 accumulators).
// All loads are unconditional with clamped addresses; masking via selects
// (v_cndmask) so EXEC never toggles around the WMMA chain.
// ---------------------------------------------------------------------------
template <int K>
__global__ void k_gemm_nodes(const float* __restrict__ X, const float* __restrict__ W,
                             float* __restrict__ H, int N) {
    constexpr int NK = (K + 3) / 4;
    int wave = threadIdx.x >> 5;
    int lane = threadIdx.x & 31;
    int tile = blockIdx.x * 4 + wave;
    int m0 = tile * 16;
    if (m0 >= N) return;                 // wave-uniform exit: EXEC stays all-1s
    int mr = lane & 15;                  // row-in-tile for A, col for B
    int kh = (lane >> 4) & 1;            // K-half select (0/1), range-provable
    int row = m0 + mr;
    bool rowok = row < N;
    int rowc = rowok ? row : (N - 1);    // clamped address, always in-bounds

    // Preload all B fragments (weights) before the WMMA chain.
    v2f Blo[NK], Bhi[NK];
#pragma unroll
    for (int s = 0; s < NK; ++s) {
        int ka = s * 4 + kh * 2;
        bool k0ok = (ka + 0) < K;
        bool k1ok = (ka + 1) < K;
        int k0c = k0ok ? (ka + 0) : 0;
        int k1c = k1ok ? (ka + 1) : 0;
        float w0l = W[k0c * HID + mr];
        float w1l = W[k1c * HID + mr];
        float w0h = W[k0c * HID + 16 + mr];
        float w1h = W[k1c * HID + 16 + mr];
        Blo[s].x = k0ok ? w0l : 0.0f;
        Blo[s].y = k1ok ? w1l : 0.0f;
        Bhi[s].x = k0ok ? w0h : 0.0f;
        Bhi[s].y = k1ok ? w1h : 0.0f;
    }

    v8f clo = {};
    v8f chi = {};
#pragma unroll
    for (int s = 0; s < NK; ++s) {
        int ka = s * 4 + kh * 2;
        bool k0ok = (ka + 0) < K;
        bool k1ok = (ka + 1) < K;
        int k0c = k0ok ? (ka + 0) : 0;
        int k1c = k1ok ? (ka + 1) : 0;
        float a0 = X[rowc * K + k0c];
        float a1 = X[rowc * K + k1c];
        v2f a;
        a.x = (rowok && k0ok) ? a0 : 0.0f;
        a.y = (rowok && k1ok) ? a1 : 0.0f;
        clo = __builtin_amdgcn_wmma_f32_16x16x4_f32(false, a, false, Blo[s], (short)0,
                                                    clo, false, false);
        chi = __builtin_amdgcn_wmma_f32_16x16x4_f32(false, a, false, Bhi[s], (short)0,
                                                    chi, false, false);
    }
#pragma unroll
    for (int r = 0; r < 8; ++r) {
        int orow = m0 + r + kh * 8;
        if (orow < N) {
            H[orow * HID + mr]      = clo[r];
            H[orow * HID + 16 + mr] = chi[r];
        }
    }
}

// ---------------------------------------------------------------------------
// Self-loop contribution doubles as zero-init: AGG = H * dinv^2
// ---------------------------------------------------------------------------
__global__ void k_selfinit(const float* __restrict__ H, const float* __restrict__ dinv,
                           float* __restrict__ AGG, int N) {
    int idx = blockIdx.x * blockDim.x + threadIdx.x;
    if (idx >= N * HID) return;
    int i = idx >> 5;
    float d = dinv[i];
    AGG[idx] = H[idx] * d * d;
}

// ---------------------------------------------------------------------------
// Edge scatter: 8 threads per edge, float4 gather + 4 scalar f32 atomics.
// h/agg buffers (12.8 MB) live in the 192 MB L2, so random traffic stays on-chip.
// ---------------------------------------------------------------------------
__global__ void k_edge_scatter(const int* __restrict__ row, const int* __restrict__ col,
                               const float* __restrict__ dinv, const float* __restrict__ H,
                               float* __restrict__ AGG, int E) {
    int idx = blockIdx.x * blockDim.x + threadIdx.x;
    int e  = idx >> 3;
    int c4 = (idx & 7) * 4;
    if (e >= E) return;
    int r = row[e];
    int c = col[e];
    float nrm = dinv[r] * dinv[c];
    const float4 h4 = *(const float4*)(H + (size_t)r * HID + c4);
    float* dst = AGG + (size_t)c * HID + c4;
    atomicAdd(dst + 0, h4.x * nrm);
    atomicAdd(dst + 1, h4.y * nrm);
    atomicAdd(dst + 2, h4.z * nrm);
    atomicAdd(dst + 3, h4.w * nrm);
}

// ---------------------------------------------------------------------------
// GraphNorm stats: block per graph (no atomics, batch is sorted)
// mean[g,c] = sum(agg+b)/cnt ; rstd[g,c] = rsqrt(var+eps),
// var = sum((x - mean*ms)^2)/cnt
// ---------------------------------------------------------------------------
__global__ void k_norm_stats(const float* __restrict__ AGG, const float* __restrict__ b,
                             const float* __restrict__ ms, const int* __restrict__ off,
                             float* __restrict__ mean, float* __restrict__ rstd) {
    int g = blockIdx.x;
    int start = off[g], end = off[g + 1];
    int c = threadIdx.x & 31, sub = threadIdx.x >> 5;
    __shared__ float red[4][HID];
    __shared__ float msh[HID];
    float cnt = fmaxf((float)(end - start), 1.0f);
    float bc = b[c];
    float acc = 0.0f;
    for (int i = start + sub; i < end; i += 4) acc += AGG[i * HID + c] + bc;
    red[sub][c] = acc;
    __syncthreads();
    if (sub == 0) {
        float m = (red[0][c] + red[1][c] + red[2][c] + red[3][c]) / cnt;
        mean[g * HID + c] = m;
        msh[c] = m;
    }
    __syncthreads();
    float m = msh[c];
    float msc = ms[c];
    float acc2 = 0.0f;
    for (int i = start + sub; i < end; i += 4) {
        float o = AGG[i * HID + c] + bc - m * msc;
        acc2 += o * o;
    }
    __syncthreads();
    red[sub][c] = acc2;
    __syncthreads();
    if (sub == 0) {
        float var = (red[0][c] + red[1][c] + red[2][c] + red[3][c]) / cnt;
        rstd[g * HID + c] = rsqrtf(var + EPSN);
    }
}

// ---------------------------------------------------------------------------
// Fused normalize (+bias) (+residual) (+ReLU)
// ---------------------------------------------------------------------------
__global__ void k_norm_apply(const float* __restrict__ AGG, const float* __restrict__ b,
                             const int* __restrict__ batch, const float* __restrict__ mean,
                             const float* __restrict__ rstd, const float* __restrict__ gw,
                             const float* __restrict__ be, const float* __restrict__ ms,
                             const float* __restrict__ res, float* __restrict__ dst, int N) {
    int idx = blockIdx.x * blockDim.x + threadIdx.x;
    if (idx >= N * HID) return;
    int i = idx >> 5, c = idx & 31;
    int gr = batch[i];
    float x = AGG[idx] + b[c];
    float o = x - mean[gr * HID + c] * ms[c];
    float y = gw[c] * o * rstd[gr * HID + c] + be[c];
    if (res) y += res[idx];
    dst[idx] = fmaxf(y, 0.0f);
}

// ---------------------------------------------------------------------------
// Mean pooling: block per graph
// ---------------------------------------------------------------------------
__global__ void k_pool(const float* __restrict__ X3, const int* __restrict__ off,
                       float* __restrict__ pooled) {
    int g = blockIdx.x;
    int start = off[g], end = off[g + 1];
    int c = threadIdx.x & 31, sub = threadIdx.x >> 5;
    __shared__ float red[4][HID];
    float acc = 0.0f;
    for (int i = start + sub; i < end; i += 4) acc += X3[i * HID + c];
    red[sub][c] = acc;
    __syncthreads();
    if (sub == 0) {
        float cnt = fmaxf((float)(end - start), 1.0f);
        pooled[g * HID + c] = (red[0][c] + red[1][c] + red[2][c] + red[3][c]) / cnt;
    }
}

// ---------------------------------------------------------------------------
// Final linear: OUT[G,3] = pooled[G,32] @ Wl[32,3] + bl  (WMMA, cols padded)
// Clamped-address loads + selects; no EXEC toggling inside the WMMA chain.
// ---------------------------------------------------------------------------
__global__ void k_gemm_final(const float* __restrict__ P, const float* __restrict__ Wl,
                             const float* __restrict__ bl, float* __restrict__ OUT, int G) {
    int wave = threadIdx.x >> 5;
    int lane = threadIdx.x & 31;
    int tile = blockIdx.x * 4 + wave;
    int m0 = tile * 16;
    if (m0 >= G) return;
    int mr = lane & 15;
    int kh = (lane >> 4) & 1;
    bool colok = mr < 3;
    int mrc = colok ? mr : 0;            // clamped column for Wl loads
    int row = m0 + mr;
    bool rowok = row < G;
    int rowc = rowok ? row : (G - 1);

    // Preload all B fragments (Wl padded to 16 columns with zeros).
    v2f Bm[8];
#pragma unroll
    for (int s = 0; s < 8; ++s) {
        int ka = s * 4 + kh * 2;
        float w0 = Wl[(ka + 0) * 3 + mrc];
        float w1 = Wl[(ka + 1) * 3 + mrc];
        Bm[s].x = colok ? w0 : 0.0f;
        Bm[s].y = colok ? w1 : 0.0f;
    }

    v8f c = {};
#pragma unroll
    for (int s = 0; s < 8; ++s) {
        int ka = s * 4 + kh * 2;
        float a0 = P[rowc * HID + ka + 0];
        float a1 = P[rowc * HID + ka + 1];
        v2f a;
        a.x = rowok ? a0 : 0.0f;
        a.y = rowok ? a1 : 0.0f;
        c = __builtin_amdgcn_wmma_f32_16x16x4_f32(false, a, false, Bm[s], (short)0, c,
                                                  false, false);
    }
#pragma unroll
    for (int r = 0; r < 8; ++r) {
        int grow = m0 + r + kh * 8;
        if (grow < G && colok) OUT[grow * 3 + mr] = c[r] + bl[mr];
    }
}

// ---------------------------------------------------------------------------
// Host launcher
// ---------------------------------------------------------------------------
extern "C" void kernel_launch(void* const* d_in, const int* in_sizes, int n_in,
                              void* d_out, int out_size, void* d_ws, size_t ws_size,
                              hipStream_t stream) {
    const float* x     = (const float*)d_in[0];   // [N,3]
    const int*   eidx  = (const int*)d_in[1];     // [2,E]
    const int*   batch = (const int*)d_in[2];     // [N]
    const float* W1 = (const float*)d_in[3];  const float* b1 = (const float*)d_in[4];
    const float* W2 = (const float*)d_in[5];  const float* b2 = (const float*)d_in[6];
    const float* W3 = (const float*)d_in[7];  const float* b3 = (const float*)d_in[8];
    const float* g1 = (const float*)d_in[9];  const float* be1 = (const float*)d_in[10];
    const float* ms1 = (const float*)d_in[11];
    const float* g2 = (const float*)d_in[12]; const float* be2 = (const float*)d_in[13];
    const float* ms2 = (const float*)d_in[14];
    const float* g3 = (const float*)d_in[15]; const float* be3 = (const float*)d_in[16];
    const float* ms3 = (const float*)d_in[17];
    const float* Wl = (const float*)d_in[18]; const float* bl = (const float*)d_in[19];
    float* out = (float*)d_out;

    const int N = in_sizes[0] / 3;
    const int E = in_sizes[1] / 2;
    const int G = out_size / 3;
    const int* erow = eidx;
    const int* ecol = eidx + E;

    // Workspace layout (~39 MB)
    float* dinv   = (float*)d_ws;                     // N
    float* B1     = dinv + N;                         // N*HID
    float* B2     = B1 + (size_t)N * HID;             // N*HID
    float* B3     = B2 + (size_t)N * HID;             // N*HID
    float* meanb  = B3 + (size_t)N * HID;             // G*HID
    float* rstdb  = meanb + (size_t)G * HID;          // G*HID
    float* pooled = rstdb + (size_t)G * HID;          // G*HID
    int*   off    = (int*)(pooled + (size_t)G * HID); // G+1

    const int T = 256;
    auto cdiv = [](int a, int b) { return (a + b - 1) / b; };

    // degree -> dinv
    k_init_deg<<<cdiv(N, T), T, 0, stream>>>(dinv, N);
    k_deg_edges<<<cdiv(E, T), T, 0, stream>>>(ecol, dinv, E);
    k_dinv<<<cdiv(N, T), T, 0, stream>>>(dinv, N);

    // per-graph node ranges
    k_graph_offsets<<<cdiv(N, T), T, 0, stream>>>(batch, off, N, G);

    const int gemmBlocks = cdiv(cdiv(N, 16), 4);   // 4 waves/block, 16 rows/wave
    const int scatBlocks = cdiv(E * 8, T);
    const int elemBlocks = cdiv(N * HID, T);

    // ---- Layer 1: x[N,3] -> x1 (B3) ----
    k_gemm_nodes<3><<<gemmBlocks, 128, 0, stream>>>(x, W1, B1, N);
    k_selfinit<<<elemBlocks, T, 0, stream>>>(B1, dinv, B2, N);
    k_edge_scatter<<<scatBlocks, T, 0, stream>>>(erow, ecol, dinv, B1, B2, E);
    k_norm_stats<<<G, 128, 0, stream>>>(B2, b1, ms1, off, meanb, rstdb);
    k_norm_apply<<<elemBlocks, T, 0, stream>>>(B2, b1, batch, meanb, rstdb, g1, be1, ms1,
                                               nullptr, B3, N);

    // ---- Layer 2: x1 (B3) -> x2 (B1), residual x1 ----
    k_gemm_nodes<32><<<gemmBlocks, 128, 0, stream>>>(B3, W2, B1, N);
    k_selfinit<<<elemBlocks, T, 0, stream>>>(B1, dinv, B2, N);
    k_edge_scatter<<<scatBlocks, T, 0, stream>>>(erow, ecol, dinv, B1, B2, E);
    k_norm_stats<<<G, 128, 0, stream>>>(B2, b2, ms2, off, meanb, rstdb);
    k_norm_apply<<<elemBlocks, T, 0, stream>>>(B2, b2, batch, meanb, rstdb, g2, be2, ms2,
                                               B3, B1, N);

    // ---- Layer 3: x2 (B1) -> x3 (B3), residual x2 ----
    k_gemm_nodes<32><<<gemmBlocks, 128, 0, stream>>>(B1, W3, B3, N);
    k_selfinit<<<elemBlocks, T, 0, stream>>>(B3, dinv, B2, N);
    k_edge_scatter<<<scatBlocks, T, 0, stream>>>(erow, ecol, dinv, B3, B2, E);
    k_norm_stats<<<G, 128, 0, stream>>>(B2, b3, ms3, off, meanb, rstdb);
    k_norm_apply<<<elemBlocks, T, 0, stream>>>(B2, b3, batch, meanb, rstdb, g3, be3, ms3,
                                               B1, B3, N);

    // ---- Pool + final linear ----
    k_pool<<<G, 128, 0, stream>>>(B3, off, pooled);
    k_gemm_final<<<cdiv(cdiv(G, 16), 4), 128, 0, stream>>>(pooled, Wl, bl, out, G);
}